// SAGE_58299886076120
// MI455X (gfx1250) — compile-verified
//
#include <hip/hip_runtime.h>

typedef __attribute__((ext_vector_type(2))) float v2f;
typedef __attribute__((ext_vector_type(8))) float v8f;

#define N_NODES 50000
#define N_EDGES 800000
#define FEAT    128
#define BATCH   2

// ---------------------------------------------------------------------------
// Degree accumulation: one thread per edge, f32 atomic into deg[dst].
// ---------------------------------------------------------------------------
__global__ void sage_deg_kernel(const int* __restrict__ dst, float* __restrict__ deg) {
    int e = blockIdx.x * blockDim.x + threadIdx.x;
    if (e < N_EDGES) atomicAdd(&deg[dst[e]], 1.0f);
}

__global__ void sage_deginv_kernel(float* __restrict__ deg) {
    int i = blockIdx.x * blockDim.x + threadIdx.x;
    if (i < N_NODES) {
        float d = deg[i];
        deg[i] = d > 0.0f ? 1.0f / d : 0.0f;   // DGL convention: 0 for isolated nodes
    }
}

// ---------------------------------------------------------------------------
// Edge scatter: agg[dst] += h[src]. 32 lanes cover one edge's 128 features
// (float4 per lane) -> coalesced gather from h, 4 f32 atomics per lane.
// Both h and agg are 25.6 MB -> resident in the 192 MB L2.
// ---------------------------------------------------------------------------
__global__ __launch_bounds__(256)
void sage_scatter_kernel(const float* __restrict__ h,
                         const int* __restrict__ src,
                         const int* __restrict__ dst,
                         float* __restrict__ agg) {
    long long idx = (long long)blockIdx.x * blockDim.x + threadIdx.x;
    int       c   = (int)(idx & 31);   // feature chunk of 4 floats
    long long e   = idx >> 5;
    if (e >= N_EDGES) return;
    int s = src[e];
    int d = dst[e];
    float4 v = ((const float4*)(h + (long long)s * FEAT))[c];
    float* ap = agg + (long long)d * FEAT + c * 4;
    atomicAdd(ap + 0, v.x);
    atomicAdd(ap + 1, v.y);
    atomicAdd(ap + 2, v.z);
    atomicAdd(ap + 3, v.w);
}

// ---------------------------------------------------------------------------
// Fused SAGE layer GEMM: out = relu(h @ Wself^T + (agg*deg_inv) @ Wneigh^T + b)
// One wave computes one 16x16 (node x outfeat) tile with V_WMMA_F32_16X16X4_F32.
// Block = 256 threads = 8 waves = the 8 output-feature tiles of one node tile.
//
// ISA 16x4 f32 A layout: lane L holds row M=L%16, K-pair base = (L<16?0:2),
// i.e. {a.x,a.y} = A[M][kbase], A[M][kbase+1]  -> contiguous 8B load.
// B (4x16, K x N) mirrors this: lane L holds col N=L%16, same K-pair.
// C/D: VGPR i -> row i + (L<16?0:8), col L%16.
// ---------------------------------------------------------------------------
__global__ __launch_bounds__(256)
void sage_gemm_kernel(const float* __restrict__ h,
                      const float* __restrict__ agg,
                      const float* __restrict__ deg_inv,
                      const float* __restrict__ Wself,
                      const float* __restrict__ Wneigh,
                      const float* __restrict__ bias,
                      float* __restrict__ out) {
    const int wave  = threadIdx.x >> 5;   // 0..7 : output-feature tile
    const int lane  = threadIdx.x & 31;
    const int half  = lane >> 4;          // 0/1 : which K-pair / which row half
    const int l15   = lane & 15;
    const int node0 = blockIdx.x * 16;    // 3125 blocks * 16 == 50000 exactly
    const int n0    = wave * 16;
    const int row   = node0 + l15;        // A-matrix row for this lane
    const int col   = n0 + l15;           // B-matrix col for this lane
    const int koff  = half * 2;

    const float* hrow  = h      + (long long)row * FEAT;
    const float* arow  = agg    + (long long)row * FEAT;
    const float* wsrow = Wself  + (long long)col * FEAT;  // B[k][n] = W[n][k]
    const float* wnrow = Wneigh + (long long)col * FEAT;
    const float  dinv  = deg_inv[row];

    v8f c = {};
#pragma unroll
    for (int kk = 0; kk < FEAT / 4; ++kk) {
        const int k = kk * 4 + koff;      // even -> 8B aligned
        v2f a_s = *(const v2f*)(hrow  + k);
        v2f b_s = *(const v2f*)(wsrow + k);
        c = __builtin_amdgcn_wmma_f32_16x16x4_f32(
                false, a_s, false, b_s, (short)0, c, false, false);
        v2f a_n = *(const v2f*)(arow  + k);
        a_n.x *= dinv;
        a_n.y *= dinv;                    // fold mean-normalization into A
        v2f b_n = *(const v2f*)(wnrow + k);
        c = __builtin_amdgcn_wmma_f32_16x16x4_f32(
                false, a_n, false, b_n, (short)0, c, false, false);
    }

    const float bv = bias[col];
#pragma unroll
    for (int i = 0; i < 8; ++i) {
        const int r = i + half * 8;
        float v = c[i] + bv;
        v = v > 0.0f ? v : 0.0f;          // ReLU
        out[(long long)(node0 + r) * FEAT + col] = v;
    }
}

// ---------------------------------------------------------------------------
// Host orchestration (graph-capture safe: only async ops on `stream`).
// Workspace layout (floats): deg/deg_inv [N] | agg [N*128] | bufA | bufB
// Total = 4*(N + 3*N*128) bytes ~= 77 MB.
// ---------------------------------------------------------------------------
extern "C" void kernel_launch(void* const* d_in, const int* in_sizes, int n_in,
                              void* d_out, int out_size, void* d_ws, size_t ws_size,
                              hipStream_t stream) {
    (void)in_sizes; (void)n_in; (void)out_size; (void)ws_size;

    const float* inputs = (const float*)d_in[0];
    const int*   src    = (const int*)d_in[1];
    const int*   dst    = (const int*)d_in[2];
    const float* Wsl[3] = { (const float*)d_in[3], (const float*)d_in[6], (const float*)d_in[9]  };
    const float* Wng[3] = { (const float*)d_in[4], (const float*)d_in[7], (const float*)d_in[10] };
    const float* bv [3] = { (const float*)d_in[5], (const float*)d_in[8], (const float*)d_in[11] };
    float* out = (float*)d_out;

    float* ws   = (float*)d_ws;
    float* deg  = ws;                                  // N, becomes deg_inv
    float* agg  = deg  + N_NODES;                      // N*128
    float* bufA = agg  + (size_t)N_NODES * FEAT;       // N*128
    float* bufB = bufA + (size_t)N_NODES * FEAT;       // N*128

    // Degrees (dst is batch-invariant -> compute once).
    hipMemsetAsync(deg, 0, N_NODES * sizeof(float), stream);
    sage_deg_kernel<<<(N_EDGES + 255) / 256, 256, 0, stream>>>(dst, deg);
    sage_deginv_kernel<<<(N_NODES + 255) / 256, 256, 0, stream>>>(deg);

    const long long scat_threads = (long long)N_EDGES * 32;
    const int scat_blocks = (int)((scat_threads + 255) / 256);
    const int gemm_blocks = N_NODES / 16;              // 3125, exact cover

    for (int b = 0; b < BATCH; ++b) {
        const float* cur = inputs + (size_t)b * N_NODES * FEAT;
        float* layer_out[3] = { bufA, bufB, out + (size_t)b * N_NODES * FEAT };
        for (int l = 0; l < 3; ++l) {
            hipMemsetAsync(agg, 0, (size_t)N_NODES * FEAT * sizeof(float), stream);
            sage_scatter_kernel<<<scat_blocks, 256, 0, stream>>>(cur, src, dst, agg);
            sage_gemm_kernel<<<gemm_blocks, 256, 0, stream>>>(
                cur, agg, deg, Wsl[l], Wng[l], bv[l], layer_out[l]);
            cur = layer_out[l];
        }
    }
}